// SNN_15504831939373
// MI455X (gfx1250) — compile-verified
//
#include <hip/hip_runtime.h>
#include <hip/hip_bf16.h>

// Problem constants (from reference): B=16, T=128, H=256, N=64, D_BERT=768
#define B_    16
#define T_    128
#define H_    256
#define N_    64
#define DB_   768
#define STEPS (T_ * H_)   // 32768 sequential LIF steps per block
#define ROWS  (B_ * T_)   // 2048 rows for the GEMMs

typedef float v2f __attribute__((ext_vector_type(2)));
typedef float v8f __attribute__((ext_vector_type(8)));

__device__ __forceinline__ float sigmoidf_(float x) {
    return 1.0f / (1.0f + __expf(-x));
}

// ---------------------------------------------------------------------------
// GEMM (f32, exact) via V_WMMA_F32_16X16X4_F32, fused bias + sigmoid epilogue.
// One wave computes a 16x64 output tile: one shared A fragment feeds FOUR
// independent WMMA accumulation chains (4 N-subtiles). This cuts A traffic 4x
// and gives the scheduler 4 independent wmma dep-chains per k-group to hide
// the WMMA latency (the f32 op is only K=4 deep).
//
// Fragment layouts per CDNA5 ISA 7.12.2:
//   A 16x4 : lane l -> row m=l&15; VGPR0 = K=2*(l>>4), VGPR1 = K=2*(l>>4)+1
//   B 4x16 : lane l -> col n=l&15; VGPR0 = row K=2*(l>>4), VGPR1 = next row
//   C 16x16: VGPR v, lanes 0-15 -> (M=v,   N=lane),
//                    lanes16-31 -> (M=v+8, N=lane-16)
// ---------------------------------------------------------------------------
__global__ __launch_bounds__(32) void gemm_sig_wmma(
    const float* __restrict__ A, const float* __restrict__ W,
    const float* __restrict__ bias, float* __restrict__ out, int K)
{
    const int lane = threadIdx.x;
    const int half = lane >> 4;      // 0: lanes 0-15, 1: lanes 16-31
    const int l15  = lane & 15;
    const int tn   = blockIdx.x;     // 64-col tile (0..3)
    const int tm   = blockIdx.y;     // 16-row tile (0..127)

    const int arow = tm * 16 + l15;
    const int col0 = tn * 64 + l15;

    const float* Ap  = A + (size_t)arow * K + 2 * half;      // 8B aligned
    const float* Wp0 = W + (size_t)(2 * half) * H_ + col0;   // q-th frag: +16*q

    v8f c0 = {}, c1 = {}, c2 = {}, c3 = {};
    for (int k = 0; k < K; k += 4) {
        v2f a = *reinterpret_cast<const v2f*>(Ap + k);       // A[m][k+2h, k+2h+1]
        const float* wk = Wp0 + (size_t)k * H_;
        v2f b0, b1, b2, b3;
        b0.x = wk[ 0];      b0.y = wk[ 0 + H_];
        b1.x = wk[16];      b1.y = wk[16 + H_];
        b2.x = wk[32];      b2.y = wk[32 + H_];
        b3.x = wk[48];      b3.y = wk[48 + H_];
        c0 = __builtin_amdgcn_wmma_f32_16x16x4_f32(false, a, false, b0, (short)0, c0, false, false);
        c1 = __builtin_amdgcn_wmma_f32_16x16x4_f32(false, a, false, b1, (short)0, c1, false, false);
        c2 = __builtin_amdgcn_wmma_f32_16x16x4_f32(false, a, false, b2, (short)0, c2, false, false);
        c3 = __builtin_amdgcn_wmma_f32_16x16x4_f32(false, a, false, b3, (short)0, c3, false, false);
    }

    const int rbase = tm * 16 + half * 8;
    float* orow = out + (size_t)rbase * H_ + col0;
#pragma unroll
    for (int q = 0; q < 4; ++q) {
        const v8f& c = (q == 0) ? c0 : (q == 1) ? c1 : (q == 2) ? c2 : c3;
        const float bv = bias[col0 + 16 * q];
#pragma unroll
        for (int v = 0; v < 8; ++v) {
            orow[(size_t)v * H_ + 16 * q] = sigmoidf_(c[v] + bv);
        }
    }
}

// ---------------------------------------------------------------------------
// LIF scan. The acc[N] channels are independent: wave w owns channels
// n0=2w (lanes 0-15 = batch) and n1=2w+1 (lanes 16-31 = batch). The per-step
// max over batch is 4x shfl_xor inside each 16-lane half -> no barriers at
// all across the 32768-step recurrence; 32 independent waves run the 64
// channels. Steps processed 4 at a time so all I/O is coalesced float4
// (j is the fastest axis of both sig and the spike tensor; 256 % 4 == 0 so
// a 4-group never crosses a t boundary).
//
//   a        = acc[n] + sigmoid_x[b, step]
//   spike    = a > thresh[n]
//   contrib  = spike ? sigmoid(a) : 0.5          (sigmoid(out), out=0 if !spike)
//   acc[n]  <- max_b (spike ? 0 : a)
//
// spikes: [B,T,N,H] floats (0/1). partial: [32][B][STEPS] per-wave pair-sums
// of contrib over its 2 channels (deterministic tree reduce afterwards).
// ---------------------------------------------------------------------------
__global__ __launch_bounds__(32) void lif_scan(
    const float* __restrict__ sig, const float* __restrict__ thresh,
    float* __restrict__ spikes, float* __restrict__ partial)
{
    const int w    = blockIdx.x;          // 0..31
    const int lane = threadIdx.x;
    const int b    = lane & 15;
    const int half = lane >> 4;
    const int n    = 2 * w + half;

    const float th = thresh[n];
    const float* xrow = sig + (size_t)b * STEPS;                  // [b][t*H+j]
    float* prow = partial + (size_t)w * (B_ * STEPS) + (size_t)b * STEPS;
    const size_t base_bn = (size_t)b * ((size_t)T_ * N_ * H_) + (size_t)n * H_;

    float acc = 0.0f;
    for (int s0 = 0; s0 < STEPS; s0 += 4) {
        // stream the L2-resident sig row ahead of the serial recurrence
        __builtin_prefetch(xrow + s0 + 64, 0, 1);
        float4 x4 = *reinterpret_cast<const float4*>(xrow + s0);
        float xs[4] = {x4.x, x4.y, x4.z, x4.w};
        float sp[4], ct[4];
#pragma unroll
        for (int i = 0; i < 4; ++i) {
            float a   = acc + xs[i];
            bool  spk = a > th;
            sp[i] = spk ? 1.0f : 0.0f;
            ct[i] = spk ? sigmoidf_(a) : 0.5f;
            float v = spk ? 0.0f : a;
            // max over the 16 batch lanes of this channel's half-wave
            v = fmaxf(v, __shfl_xor(v, 1, 32));
            v = fmaxf(v, __shfl_xor(v, 2, 32));
            v = fmaxf(v, __shfl_xor(v, 4, 32));
            v = fmaxf(v, __shfl_xor(v, 8, 32));
            acc = v;
        }
        const int t  = s0 >> 8;        // / H_
        const int j0 = s0 & (H_ - 1);
        // spike writes: contiguous in j -> 16B stores
        *reinterpret_cast<float4*>(spikes + base_bn + (size_t)t * (N_ * H_) + j0)
            = make_float4(sp[0], sp[1], sp[2], sp[3]);
        // pair-sum the two channels of this wave, lanes 0-15 store
        float p0 = ct[0] + __shfl_xor(ct[0], 16, 32);
        float p1 = ct[1] + __shfl_xor(ct[1], 16, 32);
        float p2 = ct[2] + __shfl_xor(ct[2], 16, 32);
        float p3 = ct[3] + __shfl_xor(ct[3], 16, 32);
        if (half == 0) {
            *reinterpret_cast<float4*>(prow + s0) = make_float4(p0, p1, p2, p3);
        }
    }
}

// Deterministic reduction of the 32 per-wave partial buffers into
// h_int[b][t*H+j] (== [B,T,H] row-major, directly the next GEMM's A).
__global__ void reduce_partial(const float* __restrict__ partial,
                               float* __restrict__ hint)
{
    size_t i = (size_t)blockIdx.x * blockDim.x + threadIdx.x;  // 0..524287
    float s = 0.0f;
#pragma unroll
    for (int w = 0; w < 32; ++w)
        s += partial[(size_t)w * (B_ * STEPS) + i];
    hint[i] = s;
}

// Pooler: pooled[b,t] = tanh(h @ Wp + bp)
__global__ void pooler_k(const float* __restrict__ h, const float* __restrict__ Wp,
                         const float* __restrict__ bp, float* __restrict__ out)
{
    int r = blockIdx.x * blockDim.x + threadIdx.x;
    if (r >= ROWS) return;
    float s = bp[0];
#pragma unroll 8
    for (int k = 0; k < H_; ++k)
        s += h[(size_t)r * H_ + k] * Wp[k];
    out[r] = tanhf(s);
}

extern "C" void kernel_launch(void* const* d_in, const int* in_sizes, int n_in,
                              void* d_out, int out_size, void* d_ws, size_t ws_size,
                              hipStream_t stream)
{
    const float* inputs = (const float*)d_in[0];
    const float* W1 = (const float*)d_in[1];
    const float* b1 = (const float*)d_in[2];
    const float* t1 = (const float*)d_in[3];
    const float* W2 = (const float*)d_in[4];
    const float* b2 = (const float*)d_in[5];
    const float* t2 = (const float*)d_in[6];
    const float* W3 = (const float*)d_in[7];
    const float* b3 = (const float*)d_in[8];
    const float* t3 = (const float*)d_in[9];
    const float* Wp = (const float*)d_in[10];
    const float* bp = (const float*)d_in[11];

    float* pooled = (float*)d_out;                 // [B,T] = 2048
    float* spikes = pooled + ROWS;                 // [3,B,T,N,H] as 0/1 floats
    const size_t SPK_L = (size_t)B_ * T_ * N_ * H_;  // 33,554,432 per level

    // Workspace: sig (2MB) | h_int (2MB) | partial (64MB)
    float* sig     = (float*)d_ws;
    float* hint    = sig  + (size_t)ROWS * H_;
    float* partial = hint + (size_t)ROWS * H_;

    const dim3 gemm_grid(H_ / 64, ROWS / 16);   // 4 x 128 tiles of 16x64, 1 wave each
    const dim3 wave(32);

    // ---- Block 1 ----
    gemm_sig_wmma<<<gemm_grid, wave, 0, stream>>>(inputs, W1, b1, sig, DB_);
    lif_scan<<<dim3(32), wave, 0, stream>>>(sig, t1, spikes + 0 * SPK_L, partial);
    reduce_partial<<<dim3((ROWS * H_) / 256), dim3(256), 0, stream>>>(partial, hint);

    // ---- Block 2 ----
    gemm_sig_wmma<<<gemm_grid, wave, 0, stream>>>(hint, W2, b2, sig, H_);
    lif_scan<<<dim3(32), wave, 0, stream>>>(sig, t2, spikes + 1 * SPK_L, partial);
    reduce_partial<<<dim3((ROWS * H_) / 256), dim3(256), 0, stream>>>(partial, hint);

    // ---- Block 3 ----
    gemm_sig_wmma<<<gemm_grid, wave, 0, stream>>>(hint, W3, b3, sig, H_);
    lif_scan<<<dim3(32), wave, 0, stream>>>(sig, t3, spikes + 2 * SPK_L, partial);
    reduce_partial<<<dim3((ROWS * H_) / 256), dim3(256), 0, stream>>>(partial, hint);

    // ---- Pooler ----
    pooler_k<<<dim3((ROWS + 255) / 256), dim3(256), 0, stream>>>(hint, Wp, bp, pooled);
}